// SIS_25855703122404
// MI455X (gfx1250) — compile-verified
//
#include <hip/hip_runtime.h>

// SIS trajectory: 100,000-row serial recurrence. Latency-bound on a single
// dependency chain; one wave32 computes it, LDS-stages rows, and drains them
// to HBM with the CDNA5 async LDS->global store engine (ASYNCcnt), double-
// buffered so the chain never waits on memory.

#define STEPS 100000
#define LANES 32
#define TILE_ROWS 512                         // rows per LDS buffer (4 KB)
#define ISSUES_PER_TILE (TILE_ROWS / LANES)   // 16 async b64 issues per tile

__global__ __launch_bounds__(LANES, 1)
void sis_traj_kernel(const float* __restrict__ x,
                     const float* __restrict__ beta_p,
                     const float* __restrict__ gamma_p,
                     float* __restrict__ out)
{
    __shared__ float2 buf[2][TILE_ROWS];

    const int lane = (int)threadIdx.x;

    const float S0     = x[0];
    const float I0     = x[1];
    const float beta_w = beta_p[0];
    const float gamm_w = gamma_p[0];
    const float pop    = S0 + I0;
    const float invpop = 1.0f / pop;   // hoisted; map is contractive so ulp
                                       // difference vs per-step divide decays

    float S = S0, I = I0;

    float2* gout = (float2*)out;

    const int n_full    = STEPS / TILE_ROWS;              // 195 full tiles
    const int tail_rows = STEPS - n_full * TILE_ROWS;     // 160
    const int tail_iss  = tail_rows / LANES;              // 5

    for (int t = 0; t <= n_full; ++t) {
        const int rows   = (t < n_full) ? TILE_ROWS : tail_rows;
        const int issues = (t < n_full) ? ISSUES_PER_TILE : tail_iss;
        float2* b = buf[t & 1];

        // Buffer-reuse guard: allow only the *other* buffer's async stores
        // to remain in flight (async stores complete in order).
        asm volatile("s_wait_asynccnt %0" :: "i"(ISSUES_PER_TILE) : "memory");

        if (lane == 0) {
            for (int k = 0; k < rows; ++k) {
                b[k] = make_float2(S, I);                 // ds_store_b64, off critical path
                // Reference op order:
                float infection = beta_w * S * I * invpop;  // ((b*S)*I)/pop
                float recovery  = gamm_w * I;               // g*I
                float Sn = S - infection + recovery;
                float In = I + infection - recovery;
                S = Sn;
                I = In;
            }
        }
        // Lane0's DS stores must land before the async engine reads LDS.
        asm volatile("s_wait_dscnt 0x0" ::: "memory");

        // Drain tile t in the background: 32 rows (32 lanes x b64) per issue.
        const int base_row = t * TILE_ROWS;
        for (int i = 0; i < issues; ++i) {
            const int r = i * LANES + lane;
            unsigned int       lds_addr = (unsigned int)(size_t)&b[r];
            unsigned long long gaddr    =
                (unsigned long long)(size_t)(gout + base_row + r);
            asm volatile("global_store_async_from_lds_b64 %0, %1, off"
                         :: "v"(gaddr), "v"(lds_addr) : "memory");
        }
    }

    // All async traffic retired before wave exit (S_ENDPGM also wait-idles).
    asm volatile("s_wait_asynccnt 0x0" ::: "memory");
}

extern "C" void kernel_launch(void* const* d_in, const int* in_sizes, int n_in,
                              void* d_out, int out_size, void* d_ws, size_t ws_size,
                              hipStream_t stream) {
    (void)in_sizes; (void)n_in; (void)out_size; (void)d_ws; (void)ws_size;
    const float* x       = (const float*)d_in[0];
    const float* beta_w  = (const float*)d_in[1];
    const float* gamma_w = (const float*)d_in[2];
    float* out = (float*)d_out;

    // Single wave: the whole problem is one serial dependency chain.
    sis_traj_kernel<<<1, LANES, 0, stream>>>(x, beta_w, gamma_w, out);
}